// Chamfer1DLoss_59158879535331
// MI455X (gfx1250) — compile-verified
//
#include <hip/hip_runtime.h>
#include <hip/hip_bf16.h>

// ---------------------------------------------------------------------------
// Chamfer 1D loss via V_WMMA_F32_16X16X4_F32 rank-2 trick:
//   D = A*B with A=[a|1|0|0] (16x4), B=[1^T; -b^T; 0; 0] (4x16)
//   => D[i][j] = a_i - b_j  (one WMMA computes a 16x16 pairwise-diff tile)
// rowmin_i = min_j |D[i][j]| accumulated across all b-tiles; b staged into
// LDS with GLOBAL_LOAD_ASYNC_TO_LDS_B32 (ASYNCcnt-tracked DMA).
// ---------------------------------------------------------------------------

typedef float v2f __attribute__((ext_vector_type(2)));
typedef float v8f __attribute__((ext_vector_type(8)));

#define TPB        256          // 8 waves per block (wave32)
#define CHUNK      2048         // b-values staged in LDS per stage (8 KB)
#define TILES_PER_CHUNK (CHUNK / 16)
#define BIGF       3.4e38f

// Async DMA: global -> LDS, one dword per lane. GV addressing:
//   LDS address taken from %0, global address from VGPR pair %1.
__device__ __forceinline__ void async_copy_b32(unsigned lds_off,
                                               const float* gsrc) {
    asm volatile("global_load_async_to_lds_b32 %0, %1, off"
                 :: "v"(lds_off), "v"(gsrc)
                 : "memory");
}

__device__ __forceinline__ void wait_async0() {
#if __has_builtin(__builtin_amdgcn_s_wait_asynccnt)
    __builtin_amdgcn_s_wait_asynccnt(0);
#else
    asm volatile("s_wait_asynccnt 0x0" ::: "memory");
#endif
}

__global__ void __launch_bounds__(TPB)
chamfer_dir_kernel(const float* __restrict__ a, const float* __restrict__ b,
                   int nA, int nB, float* __restrict__ accum)
{
    __shared__ float ly[CHUNK];

    const int lane        = threadIdx.x & 31;
    const int waveInBlk   = threadIdx.x >> 5;
    const int wavesPerBlk = TPB / 32;
    const int myTile      = blockIdx.x * wavesPerBlk + waveInBlk; // a-tile idx
    const int nTilesA     = nA >> 4;

    // --- A operand, loop-invariant -----------------------------------------
    // 32-bit A 16x4 layout: lanes 0-15 = {K=0,K=1}, lanes 16-31 = {K=2,K=3}.
    float aval = 0.0f;
    if (lane < 16) {
        int idx = myTile * 16 + lane;
        if (idx < nA) aval = a[idx];
    }
    v2f aop;
    aop.x = (lane < 16) ? aval : 0.0f;   // column K=0 = a_i   (K=2 = 0)
    aop.y = (lane < 16) ? 1.0f : 0.0f;   // column K=1 = 1     (K=3 = 0)

    // B row K=0 = ones (rows K=2,K=3 in high lanes = 0)
    const float bx0 = (lane < 16) ? 1.0f : 0.0f;

    // Per-lane running min of |a_i - b_j| in the 16x16 C/D layout:
    // m[v] covers row i = v + 8*(lane>=16), columns j%16 == lane&15.
    float m[8];
#pragma unroll
    for (int v = 0; v < 8; ++v) m[v] = BIGF;

    for (int base = 0; base < nB; base += CHUNK) {
        __syncthreads();                       // previous stage fully consumed

        if (base + CHUNK <= nB) {
            // Full chunk: async DMA straight into LDS (no VGPR round-trip).
#pragma unroll
            for (int k = threadIdx.x; k < CHUNK; k += TPB)
                async_copy_b32((unsigned)(size_t)&ly[k], b + base + k);
            wait_async0();                     // drain ASYNCcnt before barrier
        } else {
            // Ragged tail: scalar staging with BIGF padding (never wins min).
            for (int k = threadIdx.x; k < CHUNK; k += TPB) {
                int gi = base + k;
                ly[k] = (gi < nB) ? b[gi] : BIGF;
            }
        }
        __syncthreads();

        // Constant trip count: padded entries are BIGF, harmless under min.
#pragma unroll 4
        for (int tt = 0; tt < TILES_PER_CHUNK; ++tt) {
            float bv = ly[tt * 16 + (lane & 15)];
            v2f bop;
            bop.x = bx0;                              // B[0][j] = 1
            bop.y = (lane < 16) ? -bv : 0.0f;         // B[1][j] = -b_j
            v8f c = {};
            // D[i][j] = a_i - b_j  (EXEC all-1s: wave-uniform control flow)
            v8f d = __builtin_amdgcn_wmma_f32_16x16x4_f32(
                        false, aop, false, bop, (short)0, c, false, false);
#pragma unroll
            for (int v = 0; v < 8; ++v)
                m[v] = fminf(m[v], fabsf(d[v]));      // v_min3 with |.| mods
        }
    }

    // --- Reduce: min over the 16 lanes of each half (rows of the tile) -----
#pragma unroll
    for (int off = 1; off <= 8; off <<= 1) {
#pragma unroll
        for (int v = 0; v < 8; ++v)
            m[v] = fminf(m[v], __shfl_xor(m[v], off, 32));
    }
    // Every lane in each half now holds rowmin for its 8 rows; sum them.
    float s = 0.0f;
#pragma unroll
    for (int v = 0; v < 8; ++v) s += m[v];
    s += __shfl_xor(s, 16, 32);                      // combine both halves

    if (lane == 0 && myTile < nTilesA)
        atomicAdd(accum, s);
}

__global__ void chamfer_init_kernel(float* ws)
{
    if (threadIdx.x < 2 && blockIdx.x == 0) ws[threadIdx.x] = 0.0f;
}

__global__ void chamfer_final_kernel(const float* __restrict__ ws,
                                     float* __restrict__ out, int n, int m)
{
    if (threadIdx.x == 0 && blockIdx.x == 0) {
        const float alpha = 0.5f;
        out[0] = alpha * ws[0] / (float)n + (1.0f - alpha) * ws[1] / (float)m;
    }
}

extern "C" void kernel_launch(void* const* d_in, const int* in_sizes, int n_in,
                              void* d_out, int out_size, void* d_ws, size_t ws_size,
                              hipStream_t stream)
{
    (void)n_in; (void)out_size; (void)ws_size;
    const float* x = (const float*)d_in[0];
    const float* y = (const float*)d_in[1];
    const int n = in_sizes[0];
    const int m = in_sizes[1];
    float* ws = (float*)d_ws;

    chamfer_init_kernel<<<1, 32, 0, stream>>>(ws);

    const int tilesX = (n + 15) / 16;
    const int tilesY = (m + 15) / 16;
    const int wavesPerBlk = TPB / 32;

    // direction 1: sum_i min_j |x_i - y_j|
    chamfer_dir_kernel<<<(tilesX + wavesPerBlk - 1) / wavesPerBlk, TPB, 0, stream>>>(
        x, y, n, m, &ws[0]);
    // direction 2: sum_j min_i |y_j - x_i|
    chamfer_dir_kernel<<<(tilesY + wavesPerBlk - 1) / wavesPerBlk, TPB, 0, stream>>>(
        y, x, m, n, &ws[1]);

    chamfer_final_kernel<<<1, 32, 0, stream>>>(ws, (float*)d_out, n, m);
}